// AverageDistanceLoss_23940147708583
// MI455X (gfx1250) — compile-verified
//
#include <hip/hip_runtime.h>
#include <hip/hip_bf16.h>

typedef __attribute__((ext_vector_type(2))) float v2f;
typedef __attribute__((ext_vector_type(8))) float v8f;

#define ADL_MARGIN 0.01f
#define MAXP 1024

__device__ __forceinline__ void quat_to_R(const float q[4], float R[9]) {
    float s = q[0], u = q[1], v = q[2], w = q[3];
    R[0] = 1.f - 2.f * (v * v + w * w); R[1] = 2.f * (u * v - s * w);       R[2] = 2.f * (u * w + s * v);
    R[3] = 2.f * (u * v + s * w);       R[4] = 1.f - 2.f * (u * u + w * w); R[5] = 2.f * (v * w - s * u);
    R[6] = 2.f * (u * w - s * v);       R[7] = 2.f * (v * w + s * u);       R[8] = 1.f - 2.f * (u * u + v * v);
}

// One block per batch element b. 256 threads = 8 wave32s.
__global__ __launch_bounds__(256) void adl_main(
    const float* __restrict__ pred,    // [B,C,4]
    const float* __restrict__ targ,    // [B,C,4]
    const int*   __restrict__ labels,  // [B]
    const float* __restrict__ points,  // [C,P,3]
    const int*   __restrict__ sym,     // [C]
    float*       __restrict__ partial, // [B] per-block sums
    int C, int P)
{
    __shared__ float adat[MAXP][4];  // {-2*x2.x, -2*x2.y, -2*x2.z, |x2|^2}  (A rows, M=q)
    __shared__ float bdat[MAXP][4];  // { x1.x,   x1.y,   x1.z,   1.0   }    (B cols, N=p)
    __shared__ float n1s[MAXP];      // |x1|^2
    __shared__ float red[256];

    const int b   = blockIdx.x;
    const int tid = threadIdx.x;
    const int label = labels[b];

    if (label <= 0) {                 // background class: no loss (block-uniform exit)
        if (tid == 0) partial[b] = 0.f;
        return;
    }

    float R1[9], R2[9];
    {
        const float* q1p = pred + ((size_t)b * C + label) * 4;
        const float* q2p = targ + ((size_t)b * C + label) * 4;
        float q1[4] = { q1p[0], q1p[1], q1p[2], q1p[3] };
        float q2[4] = { q2p[0], q2p[1], q2p[2], q2p[3] };
        quat_to_R(q1, R1);
        quat_to_R(q2, R2);
    }
    const float* pts = points + (size_t)label * P * 3;
    const bool is_sym = sym[label] > 0;

    float acc = 0.f;

    if (!is_sym) {
        // ADD: per-point matched distance, x1-x2 = (R1-R2) @ pt. Pure elementwise.
        float M[9];
#pragma unroll
        for (int i = 0; i < 9; ++i) M[i] = R1[i] - R2[i];
        for (int p = tid; p < P; p += 256) {
            float px = pts[p * 3 + 0], py = pts[p * 3 + 1], pz = pts[p * 3 + 2];
            float dx = M[0] * px + M[1] * py + M[2] * pz;
            float dy = M[3] * px + M[4] * py + M[5] * pz;
            float dz = M[6] * px + M[7] * py + M[8] * pz;
            float d  = dx * dx + dy * dy + dz * dz;
            acc += fmaxf(0.5f * d - ADL_MARGIN, 0.f);
        }
    } else {
        // ADD-S: stage A rows / B cols / n1 in LDS.
        for (int p = tid; p < P; p += 256) {
            float px = pts[p * 3 + 0], py = pts[p * 3 + 1], pz = pts[p * 3 + 2];
            float x1x = R1[0] * px + R1[1] * py + R1[2] * pz;
            float x1y = R1[3] * px + R1[4] * py + R1[5] * pz;
            float x1z = R1[6] * px + R1[7] * py + R1[8] * pz;
            float x2x = R2[0] * px + R2[1] * py + R2[2] * pz;
            float x2y = R2[3] * px + R2[4] * py + R2[5] * pz;
            float x2z = R2[6] * px + R2[7] * py + R2[8] * pz;
            bdat[p][0] = x1x;  bdat[p][1] = x1y;  bdat[p][2] = x1z;  bdat[p][3] = 1.0f;
            adat[p][0] = -2.f * x2x; adat[p][1] = -2.f * x2y; adat[p][2] = -2.f * x2z;
            adat[p][3] = x2x * x2x + x2y * x2y + x2z * x2z;
            n1s[p] = x1x * x1x + x1y * x1y + x1z * x1z;
        }
        __syncthreads();

        const int wave = tid >> 5;
        const int lane = tid & 31;
        const int half = lane >> 4;   // lane half selects K pair (A: K=0,1 / K=2,3; B likewise)
        const int l    = lane & 15;
        const int nt   = P >> 4;      // number of 16-wide tiles

        for (int ptile = wave; ptile < nt; ptile += 8) {
            const int p = ptile * 16 + l;
            // B operand (K x N = 4 x 16): VGPR0 = K=0 (lanes 0-15) / K=2 (16-31); VGPR1 = K=1 / K=3
            v2f bOp;
            bOp.x = bdat[p][half * 2 + 0];
            bOp.y = bdat[p][half * 2 + 1];

            v8f rm;
#pragma unroll
            for (int i = 0; i < 8; ++i) rm[i] = 3.402823466e38f;

            for (int qt = 0; qt < nt; ++qt) {
                const int q = qt * 16 + l;
                // A operand (M x K = 16 x 4): VGPR0 = K=0 / K=2; VGPR1 = K=1 / K=3
                v2f aOp;
                aOp.x = adat[q][half * 2 + 0];
                aOp.y = adat[q][half * 2 + 1];
                v8f cz = { 0.f, 0.f, 0.f, 0.f, 0.f, 0.f, 0.f, 0.f };
                // D[M=q, N=p] = n2[q] - 2 * x1[p].x2[q]
                v8f d = __builtin_amdgcn_wmma_f32_16x16x4_f32(
                    false, aOp, false, bOp, (short)0, cz, false, false);
#pragma unroll
                for (int i = 0; i < 8; ++i) rm[i] = fminf(rm[i], d[i]);
            }
            // min over M (q axis): 8 accumulator VGPRs then combine lane halves.
            float m = fminf(fminf(fminf(rm[0], rm[1]), fminf(rm[2], rm[3])),
                            fminf(fminf(rm[4], rm[5]), fminf(rm[6], rm[7])));
            m = fminf(m, __shfl_xor(m, 16, 32));
            if (lane < 16) {                      // each p counted once
                float d = n1s[p] + m;
                acc += fmaxf(0.5f * d - ADL_MARGIN, 0.f);
            }
        }
    }

    // deterministic block reduction
    red[tid] = acc;
    __syncthreads();
    for (int off = 128; off > 0; off >>= 1) {
        if (tid < off) red[tid] += red[tid + off];
        __syncthreads();
    }
    if (tid == 0) partial[b] = red[0];
}

// Fixed-order final reduction (no float atomics -> bitwise deterministic).
__global__ __launch_bounds__(256) void adl_reduce(
    const float* __restrict__ partial, float* __restrict__ out, int n, float scale)
{
    __shared__ float s[256];
    int tid = threadIdx.x;
    s[tid] = (tid < n) ? partial[tid] : 0.f;
    __syncthreads();
    for (int off = 128; off > 0; off >>= 1) {
        if (tid < off) s[tid] += s[tid + off];
        __syncthreads();
    }
    if (tid == 0) out[0] = s[0] * scale;
}

extern "C" void kernel_launch(void* const* d_in, const int* in_sizes, int n_in,
                              void* d_out, int out_size, void* d_ws, size_t ws_size,
                              hipStream_t stream)
{
    const float* pred   = (const float*)d_in[0];
    const float* targ   = (const float*)d_in[1];
    const int*   labels = (const int*)d_in[2];
    const float* points = (const float*)d_in[3];
    const int*   sym    = (const int*)d_in[4];

    const int B = in_sizes[2];
    const int C = in_sizes[4];
    const int P = in_sizes[3] / (C * 3);

    float* partial = (float*)d_ws;

    adl_main<<<B, 256, 0, stream>>>(pred, targ, labels, points, sym, partial, C, P);

    const float scale = 1.0f / (float)(B * P);
    adl_reduce<<<1, 256, 0, stream>>>(partial, (float*)d_out, B, scale);
}